// Sampling_10634339025072
// MI455X (gfx1250) — compile-verified
//
#include <hip/hip_runtime.h>
#include <stdint.h>

#define BATCH 16
#define NPTS  65536
#define NSAMP 512
#define TPB   1024
#define PPT   (NPTS / TPB)   // 64 points per thread
#define CHUNKS (PPT / 4)     // 16 chunks of 4 points (3 x float4 each)
#define NWAVE (TPB / 32)     // 32 wave32s per block

// ---- CDNA5 async global<->LDS helpers (gfx1250) -------------------------
__device__ __forceinline__ unsigned lds_addr(const void* p) {
  // generic LDS pointer = {SHARED_BASE, lds_offset}; low 32 bits = LDS byte address
  return (unsigned)(uintptr_t)p;
}
__device__ __forceinline__ void async_load_b128(unsigned lds, unsigned voff,
                                                unsigned long long sbase) {
  asm volatile("global_load_async_to_lds_b128 %0, %1, %2"
               :: "v"(lds), "v"(voff), "s"(sbase) : "memory");
}
__device__ __forceinline__ void async_store_b128(unsigned voff, unsigned lds,
                                                 unsigned long long sbase) {
  asm volatile("global_store_async_from_lds_b128 %0, %1, %2"
               :: "v"(voff), "v"(lds), "s"(sbase) : "memory");
}
__device__ __forceinline__ void wait_async0() {
  asm volatile("s_wait_asynccnt 0" ::: "memory");
}

// ---- FPS: one workgroup per batch, 512 serial rounds --------------------
// Critical path per round: VALU scan -> wave32 shuffle argmax carrying the
// winner's coordinates -> LDS cross-wave reduce -> broadcast. No global
// memory access gates the serial chain.
__global__ __launch_bounds__(TPB)
void fps_kernel(const float* __restrict__ xyz, int* __restrict__ idx_out) {
  __shared__ float s_cent[3];
  __shared__ int   s_last;
  __shared__ float s_rv[NWAVE];
  __shared__ int   s_ri[NWAVE];
  __shared__ float s_rx[NWAVE];
  __shared__ float s_ry[NWAVE];
  __shared__ float s_rz[NWAVE];

  const int b    = blockIdx.x;
  const int tid  = threadIdx.x;
  const int lane = tid & 31;
  const int wv   = tid >> 5;

  const float* batch_xyz = xyz + (size_t)b * NPTS * 3;
  const float4* __restrict__ p4 =
      reinterpret_cast<const float4*>(batch_xyz + (size_t)tid * PPT * 3);

  // Seed: centroid for round 0 is point 0 of the batch.
  if (tid == 0) {
    s_cent[0] = batch_xyz[0];
    s_cent[1] = batch_xyz[1];
    s_cent[2] = batch_xyz[2];
  }

  float dist[PPT];
#pragma unroll
  for (int i = 0; i < PPT; ++i) dist[i] = 1e10f;

  int last = 0;
  __syncthreads();

  for (int s = 0; s < NSAMP; ++s) {
    if (tid == 0) idx_out[b * NSAMP + s] = last;

    const float cx = s_cent[0], cy = s_cent[1], cz = s_cent[2];

    float bv = -1.0f;
    int   bi = tid * PPT;
    float bx = 0.0f, by = 0.0f, bz = 0.0f;
#pragma unroll
    for (int c = 0; c < CHUNKS; ++c) {
      const float4 q0 = p4[c * 3 + 0];  // L2-resident, latency-hidden b128 loads
      const float4 q1 = p4[c * 3 + 1];
      const float4 q2 = p4[c * 3 + 2];
      const float px[4] = {q0.x, q0.w, q1.z, q2.y};
      const float py[4] = {q0.y, q1.x, q1.w, q2.z};
      const float pz[4] = {q0.z, q1.y, q2.x, q2.w};
#pragma unroll
      for (int j = 0; j < 4; ++j) {
        const float dx = px[j] - cx;
        const float dy = py[j] - cy;
        const float dz = pz[j] - cz;
        const float d  = dx * dx + dy * dy + dz * dz;
        const float nd = fminf(dist[c * 4 + j], d);  // dists live in VGPRs
        dist[c * 4 + j] = nd;
        if (nd > bv) {
          bv = nd; bi = tid * PPT + c * 4 + j;
          bx = px[j]; by = py[j]; bz = pz[j];
        }
      }
    }

    // wave32 argmax reduce carrying (val, idx, x, y, z); ties -> lowest idx
#pragma unroll
    for (int m = 16; m >= 1; m >>= 1) {
      const float ov = __shfl_xor(bv, m, 32);
      const int   oi = __shfl_xor(bi, m, 32);
      const float ox = __shfl_xor(bx, m, 32);
      const float oy = __shfl_xor(by, m, 32);
      const float oz = __shfl_xor(bz, m, 32);
      if (ov > bv || (ov == bv && oi < bi)) {
        bv = ov; bi = oi; bx = ox; by = oy; bz = oz;
      }
    }
    if (lane == 0) {
      s_rv[wv] = bv; s_ri[wv] = bi;
      s_rx[wv] = bx; s_ry[wv] = by; s_rz[wv] = bz;
    }
    __syncthreads();  // also fences this round's s_cent reads before overwrite

    if (wv == 0) {
      float v = s_rv[lane];
      int   i = s_ri[lane];
      float x = s_rx[lane], y = s_ry[lane], z = s_rz[lane];
#pragma unroll
      for (int m = 16; m >= 1; m >>= 1) {
        const float ov = __shfl_xor(v, m, 32);
        const int   oi = __shfl_xor(i, m, 32);
        const float ox = __shfl_xor(x, m, 32);
        const float oy = __shfl_xor(y, m, 32);
        const float oz = __shfl_xor(z, m, 32);
        if (ov > v || (ov == v && oi < i)) {
          v = ov; i = oi; x = ox; y = oy; z = oz;
        }
      }
      if (lane == 0) {
        s_last = i;
        s_cent[0] = x; s_cent[1] = y; s_cent[2] = z;  // next centroid, no mem trip
      }
    }
    __syncthreads();
    last = s_last;
  }
}

// ---- Gather: DMA-style f-row copy through LDS (CDNA5 async path) --------
__global__ __launch_bounds__(32)
void gather_kernel(const float* __restrict__ xyz, const float* __restrict__ f,
                   const int* __restrict__ idx, float* __restrict__ out_xyz,
                   float* __restrict__ out_f) {
  __shared__ float stage[64];
  const int s = blockIdx.x;
  const int b = blockIdx.y;
  const int tid = threadIdx.x;
  const int k = idx[b * NSAMP + s];

  const unsigned long long fbase = (unsigned long long)(uintptr_t)f;
  const unsigned long long obase = (unsigned long long)(uintptr_t)out_f;

  if (tid < 16) {  // 16 lanes x b128 = the 256B f-row, global -> LDS
    const unsigned voff =
        ((unsigned)(b * NPTS + k) * 64u + (unsigned)tid * 4u) * 4u;
    async_load_b128(lds_addr(stage) + 16u * (unsigned)tid, voff, fbase);
  }
  wait_async0();
  __syncthreads();
  if (tid < 16) {  // LDS -> out, no VGPR round trip
    const unsigned ooff =
        ((unsigned)(b * NSAMP + s) * 64u + (unsigned)tid * 4u) * 4u;
    async_store_b128(ooff, lds_addr(stage) + 16u * (unsigned)tid, obase);
  }
  if (tid < 3) {
    out_xyz[(b * NSAMP + s) * 3 + tid] = xyz[((size_t)b * NPTS + k) * 3 + tid];
  }
  wait_async0();  // (S_ENDPGM also implies wait-idle)
}

extern "C" void kernel_launch(void* const* d_in, const int* in_sizes, int n_in,
                              void* d_out, int out_size, void* d_ws, size_t ws_size,
                              hipStream_t stream) {
  (void)in_sizes; (void)n_in; (void)out_size; (void)ws_size;
  const float* xyz = (const float*)d_in[0];  // [16,65536,3] f32
  const float* f   = (const float*)d_in[1];  // [16,65536,64] f32
  int* idx = (int*)d_ws;                     // [16,512] selected indices
  float* out_xyz = (float*)d_out;                       // [16,512,3]
  float* out_f   = out_xyz + BATCH * NSAMP * 3;         // [16,512,64]

  fps_kernel<<<BATCH, TPB, 0, stream>>>(xyz, idx);
  gather_kernel<<<dim3(NSAMP, BATCH), 32, 0, stream>>>(xyz, f, idx, out_xyz, out_f);
}